// NLinear_17892833755539
// MI455X (gfx1250) — compile-verified
//
#include <hip/hip_runtime.h>

// ---------------------------------------------------------------------------
// NLinear: out[b,n,o] = sum_i x[b,n,i] * W[n,i,o] + bias[n,o]
// 200 independent GEMMs (M=4096, K=64, N=64), fp32 end-to-end.
// Memory-bound (~423 MB @ 23.3 TB/s ~= 18us); V_WMMA_F32_16X16X4_F32 keeps
// exact fp32 on the matrix pipe. W transposed + 68-float LDS pitch so both
// A and B fragments are single conflict-free ds_load_b64s into aligned pairs.
// ---------------------------------------------------------------------------

typedef float v2f __attribute__((ext_vector_type(2)));
typedef float v8f __attribute__((ext_vector_type(8)));
typedef int   v4i __attribute__((ext_vector_type(4)));

typedef __attribute__((address_space(1))) v4i* glob_v4i_ptr;
typedef __attribute__((address_space(3))) v4i* lds_v4i_ptr;

#define B_TOT   4096
#define N_TOK   200
#define D_IN    64
#define D_OUT   64

#define WAVES_PER_BLOCK 8
#define ROWS_PER_WAVE   16
#define ROWS_PER_BLOCK  (WAVES_PER_BLOCK * ROWS_PER_WAVE)   // 128

// LDS row pitch in floats: 68*4 B = 272 B. 272 % 16 == 0 (b128-aligned rows),
// 68 % 64 == 4 (lane stride 4 banks -> conflict-free b64 fragment reads).
#define LDS_PITCH 68

#if defined(__HIP_DEVICE_COMPILE__) && __has_builtin(__builtin_amdgcn_global_load_async_to_lds_b128)
#define USE_ASYNC_LDS 1
#else
#define USE_ASYNC_LDS 0
#endif

__global__ __launch_bounds__(256, 1) void nlinear_wmma_f32(
    const float* __restrict__ x,     // [B, N, 64]
    const float* __restrict__ w,     // [1, N, 64, 64]
    const float* __restrict__ bias,  // [1, N, 64]
    float* __restrict__ out)         // [B, N, 64]
{
    __shared__ float w_lds[D_OUT * LDS_PITCH];                      // ~17 KB, transposed: [o][i]
    __shared__ float b_lds[D_OUT];
    __shared__ float x_lds[WAVES_PER_BLOCK][ROWS_PER_WAVE * LDS_PITCH]; // ~34 KB

    const int tid  = threadIdx.x;
    const int wave = tid >> 5;            // wave32
    const int lane = tid & 31;
    const int half = lane >> 4;           // 0: lanes 0-15, 1: lanes 16-31
    const int l15  = lane & 15;

    const int n  = blockIdx.y;                                    // token position
    const int m0 = blockIdx.x * ROWS_PER_BLOCK + wave * ROWS_PER_WAVE;

    // ---- W[n] (64x64) -> LDS transposed (w_lds[o*PITCH + i]); bias[n] -> LDS ----
    {
        const float4* wsrc = (const float4*)(w + (size_t)n * D_IN * D_OUT);
        #pragma unroll
        for (int p = 0; p < 4; ++p) {          // 1024 float4 / 256 threads
            const int f  = tid + 256 * p;      // float4 index
            const int i  = f >> 4;             // K row 0..63
            const int o4 = (f & 15) * 4;       // O col
            const float4 v = wsrc[f];
            w_lds[(o4 + 0) * LDS_PITCH + i] = v.x;
            w_lds[(o4 + 1) * LDS_PITCH + i] = v.y;
            w_lds[(o4 + 2) * LDS_PITCH + i] = v.z;
            w_lds[(o4 + 3) * LDS_PITCH + i] = v.w;
        }
        if (tid < D_OUT) b_lds[tid] = bias[(size_t)n * D_OUT + tid];
    }

    // ---- stage this wave's 16x64 x-tile -> LDS (rows are N_TOK*64 floats apart) ----
    {
        const float* xbase = x + ((size_t)m0 * N_TOK + n) * D_IN;
        float*       dst   = x_lds[wave];
        #pragma unroll
        for (int p = 0; p < 8; ++p) {          // 16 rows x 256B, 1 float4/lane/iter
            const int r = 2 * p + half;        // lanes 0-15 -> row 2p, 16-31 -> row 2p+1
            const float* src = xbase + (size_t)r * (N_TOK * D_IN) + 4 * l15;
            float*       d   = dst + r * LDS_PITCH + 4 * l15;     // 16B-aligned
#if USE_ASYNC_LDS
            __builtin_amdgcn_global_load_async_to_lds_b128(
                (glob_v4i_ptr)src, (lds_v4i_ptr)d, 0, 0);
#else
            *(float4*)d = *(const float4*)src;
#endif
        }
#if USE_ASYNC_LDS
#if __has_builtin(__builtin_amdgcn_s_wait_asynccnt)
        __builtin_amdgcn_s_wait_asynccnt(0);
#else
        asm volatile("s_wait_asynccnt 0" ::: "memory");
#endif
#endif
    }
    __syncthreads();

    // ---- 16x64 output tile per wave: 4 accumulators of 16x16, K swept in steps of 4 ----
    v8f acc[4] = {v8f{}, v8f{}, v8f{}, v8f{}};

    const float* xw = x_lds[wave];

    #pragma unroll
    for (int k0 = 0; k0 < D_IN; k0 += 4) {
        const int kr = k0 + 2 * half;          // even -> 8B-aligned b64 loads
        // A frag (16x4, ISA layout): lane -> M=l15, v0/v1 hold K = kr, kr+1
        v2f a = *(const v2f*)(xw + l15 * LDS_PITCH + kr);
        #pragma unroll
        for (int t = 0; t < 4; ++t) {
            // B frag (4x16): lane -> N = 16t + l15, v0/v1 hold K = kr, kr+1
            // (single contiguous b64 from transposed W)
            v2f b = *(const v2f*)(w_lds + (16 * t + l15) * LDS_PITCH + kr);
            acc[t] = __builtin_amdgcn_wmma_f32_16x16x4_f32(
                /*neg_a=*/false, a, /*neg_b=*/false, b,
                /*c_mod=*/(short)0, acc[t], /*reuse_a=*/false, /*reuse_b=*/false);
        }
    }

    // ---- bias + writeout: C/D layout -> reg r holds M = r + 8*half, N = 16t + l15 ----
    #pragma unroll
    for (int t = 0; t < 4; ++t) {
        const float bb = b_lds[16 * t + l15];
        #pragma unroll
        for (int r = 0; r < 8; ++r) {
            const int row = m0 + r + 8 * half;
            out[((size_t)row * N_TOK + n) * D_OUT + 16 * t + l15] = acc[t][r] + bb;
        }
    }
}

extern "C" void kernel_launch(void* const* d_in, const int* in_sizes, int n_in,
                              void* d_out, int out_size, void* d_ws, size_t ws_size,
                              hipStream_t stream) {
    const float* x    = (const float*)d_in[0];
    const float* w    = (const float*)d_in[1];
    const float* bias = (const float*)d_in[2];
    float*       out  = (float*)d_out;

    dim3 grid(B_TOT / ROWS_PER_BLOCK, N_TOK);   // (32, 200)
    nlinear_wmma_f32<<<grid, 256, 0, stream>>>(x, w, bias, out);
}